// PolynomialFilter_23055384445264
// MI455X (gfx1250) — compile-verified
//
#include <hip/hip_runtime.h>

// fp32 WMMA fragment types (wave32): A(16x4)=2 VGPR, C/D(16x16)=8 VGPR
typedef __attribute__((ext_vector_type(2))) float v2f;
typedef __attribute__((ext_vector_type(8))) float v8f;

// ---------------------------------------------------------------------------
// Dense GEMM: out[N,64] (+)= X[N,64] @ W[64,64]  using V_WMMA_F32_16X16X4_F32
// Block = 256 threads = 8 waves; each wave computes a 16x64 output slab
// (four 16x16 WMMA tiles), K-loop of 16 steps (K=4 per WMMA).
//
// W is staged once per block into LDS in *B-fragment order*:
//   sW[kb/4][tile][lane] = { W[kb+2*hi][lo+16*tile], W[kb+2*hi+1][lo+16*tile] }
// with lane = hi*16+lo.  Each B fragment is then one aligned ds_load_b64 at
// base + lane*8: 32 lanes x 8B = 256B contiguous -> all 64 banks, no
// conflicts, no repacking movs in front of the WMMAs.
// ---------------------------------------------------------------------------
__global__ __launch_bounds__(256) void poly_gemm_wmma(
    const float* __restrict__ X, const float* __restrict__ W,
    const float* __restrict__ bias, float* __restrict__ out,
    int N, int init)
{
  __shared__ v2f sW[16 * 4 * 32];   // 16 KB
  for (int j = threadIdx.x; j < 16 * 4 * 32; j += 256) {
    const int kb4 = j >> 7;          // K-slab (4 wide): 0..15
    const int t   = (j >> 5) & 3;    // N tile: 0..3
    const int ln  = j & 31;          // lane position
    const int h   = ln >> 4;
    const int lw  = ln & 15;
    const int krow = kb4 * 4 + 2 * h;
    const int ncol = lw + 16 * t;
    v2f p;
    p.x = W[krow * 64 + ncol];
    p.y = W[(krow + 1) * 64 + ncol];
    sW[j] = p;
  }
  __syncthreads();

  const int lane = threadIdx.x & 31;
  const int wave = threadIdx.x >> 5;
  const int rowBase = (blockIdx.x * 8 + wave) * 16;
  if (rowBase >= N) return;

  const int lo = lane & 15;   // M (for A) / N (for B,C,D)
  const int hi = lane >> 4;   // selects K pair (A/B) or row half (C/D)

  v8f acc0 = {}, acc1 = {}, acc2 = {}, acc3 = {};
  const float* xrow = X + (size_t)(rowBase + lo) * 64;
  const v2f* wl = sW + lane;  // per-lane base into fragment-ordered W

#pragma unroll
  for (int kb = 0; kb < 64; kb += 4) {
    const int ka = kb + 2 * hi;
    // A fragment: lane holds X[row][ka], X[row][ka+1] (ISA 16x4 f32 layout)
    v2f a;
    a.x = xrow[ka];
    a.y = xrow[ka + 1];
    // B fragments: one aligned 8B LDS load each, conflict-free
    const v2f* wb = wl + (kb >> 2) * 128;
    const v2f b0 = wb[0];
    const v2f b1 = wb[32];
    const v2f b2 = wb[64];
    const v2f b3 = wb[96];
    // (neg_a, A, neg_b, B, c_mod, C, reuse_a, reuse_b)
    acc0 = __builtin_amdgcn_wmma_f32_16x16x4_f32(false, a, false, b0, (short)0, acc0, false, false);
    acc1 = __builtin_amdgcn_wmma_f32_16x16x4_f32(false, a, false, b1, (short)0, acc1, false, false);
    acc2 = __builtin_amdgcn_wmma_f32_16x16x4_f32(false, a, false, b2, (short)0, acc2, false, false);
    acc3 = __builtin_amdgcn_wmma_f32_16x16x4_f32(false, a, false, b3, (short)0, acc3, false, false);
  }

  // C/D layout: VGPR r holds row M = r + 8*hi, column N = lo
  const float bb0 = bias[lo], bb1 = bias[lo + 16],
              bb2 = bias[lo + 32], bb3 = bias[lo + 48];
#pragma unroll
  for (int r = 0; r < 8; ++r) {
    float* o = out + (size_t)(rowBase + r + 8 * hi) * 64;
    if (init) {
      o[lo]      = acc0[r] + bb0;
      o[lo + 16] = acc1[r] + bb1;
      o[lo + 32] = acc2[r] + bb2;
      o[lo + 48] = acc3[r] + bb3;
    } else {
      o[lo]      += acc0[r];
      o[lo + 16] += acc1[r];
      o[lo + 32] += acc2[r];
      o[lo + 48] += acc3[r];
    }
  }
}

// ---------------------------------------------------------------------------
// Zero a buffer (float4-vectorized)
// ---------------------------------------------------------------------------
__global__ void poly_zero(float4* __restrict__ p, int n4) {
  int i = blockIdx.x * blockDim.x + threadIdx.x;
  if (i < n4) p[i] = make_float4(0.f, 0.f, 0.f, 0.f);
}

// ---------------------------------------------------------------------------
// COO SpMM scatter: xout[row] += val * xin[col], one thread per (edge,feature)
// -> one coalesced load + one global_atomic_add_f32 per thread.
// ---------------------------------------------------------------------------
__global__ void poly_spmm(const int* __restrict__ row, const int* __restrict__ col,
                          const float* __restrict__ val,
                          const float* __restrict__ xin, float* __restrict__ xout,
                          int E)
{
  int t = blockIdx.x * blockDim.x + threadIdx.x;
  int e = t >> 6;
  int f = t & 63;
  if (e >= E) return;
  float v = val[e];
  int c = col[e];
  int r = row[e];
  atomicAdd(xout + (size_t)r * 64 + f, v * xin[(size_t)c * 64 + f]);
}

// ---------------------------------------------------------------------------
// Host launcher: out = sum_k (L^k x) W_k + bias ; K = 4
// ---------------------------------------------------------------------------
extern "C" void kernel_launch(void* const* d_in, const int* in_sizes, int n_in,
                              void* d_out, int out_size, void* d_ws, size_t ws_size,
                              hipStream_t stream) {
  const float* x       = (const float*)d_in[0];
  const int*   lap_row = (const int*)d_in[1];
  const int*   lap_col = (const int*)d_in[2];
  const float* lap_val = (const float*)d_in[3];
  const float* weights = (const float*)d_in[4];
  const float* bias    = (const float*)d_in[5];
  float* out = (float*)d_out;

  const int N = in_sizes[0] / 64;   // 50000
  const int E = in_sizes[1];        // 800000
  const int K = 4;

  float* buf[2] = { (float*)d_ws, (float*)d_ws + (size_t)N * 64 };

  const int rowBlocks = (N + 15) / 16;
  dim3 gemmGrid((rowBlocks + 7) / 8), gemmBlock(256);
  const int n4 = (N * 64) / 4;
  dim3 zeroGrid((n4 + 255) / 256), zeroBlock(256);
  const long long spmmThreads = (long long)E * 64;
  dim3 spmmGrid((unsigned)((spmmThreads + 255) / 256)), spmmBlock(256);

  const float* xk = x;
  for (int k = 0; k <= K; ++k) {
    poly_gemm_wmma<<<gemmGrid, gemmBlock, 0, stream>>>(
        xk, weights + (size_t)k * 64 * 64, bias, out, N, (k == 0) ? 1 : 0);
    if (k < K) {
      float* nxt = buf[k & 1];
      poly_zero<<<zeroGrid, zeroBlock, 0, stream>>>((float4*)nxt, n4);
      poly_spmm<<<spmmGrid, spmmBlock, 0, stream>>>(lap_row, lap_col, lap_val,
                                                    xk, nxt, E);
      xk = nxt;
    }
  }
}